// Born_35502199669494
// MI455X (gfx1250) — compile-verified
//
#include <hip/hip_runtime.h>
#include <math.h>

typedef __attribute__((ext_vector_type(2))) float v2f;
typedef __attribute__((ext_vector_type(8))) float v8f;
typedef int async_v4i __attribute__((vector_size(16)));   // matches builtin param type

#define LDP   68        // padded LDS row stride (floats): 272B rows, 16B aligned,
                        // A-frag bank = (4*row + col) % 64 -> conflict-free
#define TCUT  6.0f
#define BNEPS 1e-5f

#if defined(__HIP_DEVICE_COMPILE__) && __has_builtin(__builtin_amdgcn_global_load_async_to_lds_b128)
#define HAVE_ASYNC_LDS 1
#else
#define HAVE_ASYNC_LDS 0
#endif

__device__ __forceinline__ float softplusf(float x) {
    // stable log1p(exp(x)) = max(x,0) + log1p(exp(-|x|))
    return fmaxf(x, 0.0f) + log1pf(expf(-fabsf(x)));
}

__device__ __forceinline__ void wait_async_lds() {
#if HAVE_ASYNC_LDS
#if __has_builtin(__builtin_amdgcn_s_wait_asynccnt)
    __builtin_amdgcn_s_wait_asynccnt(0);
#else
    asm volatile("s_wait_asynccnt 0" ::: "memory");
#endif
#endif
}

// One 32x64 <- 32x64 @ 64x64 layer with bias + softplus, via fp32 WMMA.
// 8 waves: row-group = wave>>2 (16 rows), col-tile = wave&3 (16 cols).
// A (16x4 f32) from LDS: lanes 0-15 K=0,1 ; lanes 16-31 K=2,3 (M = lane&15).
// B (4x16 f32) from GLOBAL fragment-packed weights: WF[p][n] = (W[2p][n], W[2p+1][n]),
//   lane loads pair p = 2*kk + (lane>>4) at n = n0 + (lane&15)  -> one b64, coalesced.
// C/D (16x16 f32): VGPR i -> M = i + 8*(lane>=16), N = lane&15.
__device__ __forceinline__ void wmma_layer(const float* __restrict__ sIn,
                                           const v2f* __restrict__ WF,
                                           const float* __restrict__ bg,
                                           float* __restrict__ sOut)
{
    const int lane = threadIdx.x & 31;
    const int wave = threadIdx.x >> 5;
    const int r0 = (wave >> 2) << 4;     // 0 or 16
    const int n0 = (wave & 3) << 4;      // 0,16,32,48
    const int mn = lane & 15;
    const int kh = (lane >> 4) << 1;     // 0 or 2

    v8f acc = {0.f, 0.f, 0.f, 0.f, 0.f, 0.f, 0.f, 0.f};
#pragma unroll
    for (int kk = 0; kk < 16; ++kk) {
        const int kb = kk * 4 + kh;
        v2f a, b;
        a.x = sIn[(r0 + mn) * LDP + kb];
        a.y = sIn[(r0 + mn) * LDP + kb + 1];
        b = WF[(2 * kk + (lane >> 4)) * 64 + n0 + mn];
        acc = __builtin_amdgcn_wmma_f32_16x16x4_f32(false, a, false, b,
                                                    (short)0, acc, false, false);
    }
    const int drow = r0 + ((lane >> 4) << 3);
    const int dcol = n0 + mn;
    const float bias = bg[dcol];
#pragma unroll
    for (int i = 0; i < 8; ++i)
        sOut[(drow + i) * LDP + dcol] = softplusf(acc[i] + bias);
}

// Shared MLP body: async-stage a 32-row tile into LDS, two WMMA layers
// (weights direct from global, fragment-packed), then the 64->1 output dot.
__device__ __forceinline__ void mlp_core(const float* __restrict__ attr,
                                         const v2f* __restrict__ WF0,
                                         const float* __restrict__ b0,
                                         const v2f* __restrict__ WF1,
                                         const float* __restrict__ b1,
                                         const float* __restrict__ Wo,
                                         const float* __restrict__ bo,
                                         int row0, int rows, float* sRow)
{
    __shared__ float sX[32 * LDP];
    __shared__ float sH[32 * LDP];

    const int tid = threadIdx.x;
#if HAVE_ASYNC_LDS
    // 32 rows x 64 floats = 512 float4 transfers; 256 threads x 2.
#pragma unroll
    for (int i = 0; i < 2; ++i) {
        const int f   = tid + i * 256;        // float4 index
        const int r   = f >> 4;               // 16 float4 per row
        const int col = (f & 15) << 2;
        int gr = row0 + r; if (gr >= rows) gr = rows - 1;   // clamp: stay in-bounds
        const float* gsrc = attr + (size_t)gr * 64 + col;
        float* ldst = &sX[r * LDP + col];
        __builtin_amdgcn_global_load_async_to_lds_b128(
            (__attribute__((address_space(1))) async_v4i*)gsrc,
            (__attribute__((address_space(3))) async_v4i*)ldst, 0, 0);
    }
    wait_async_lds();
#else
    for (int i = tid; i < 32 * 64; i += 256) {
        const int r = i >> 6, c = i & 63;
        int gr = row0 + r; if (gr >= rows) gr = rows - 1;
        sX[r * LDP + c] = attr[(size_t)gr * 64 + c];
    }
#endif
    __syncthreads();
    wmma_layer(sX, WF0, b0, sH);      // layer 1: X -> H
    __syncthreads();
    wmma_layer(sH, WF1, b1, sX);      // layer 2: H -> X (reuse)
    __syncthreads();
    if (tid < 32) {
        float s = 0.f;
#pragma unroll
        for (int c = 0; c < 64; ++c) s += sX[tid * LDP + c] * Wo[c];
        sRow[tid] = s + bo[0];
    }
    __syncthreads();
}

// ---------------- kernels ----------------

// Fold BN into Linear and re-pack weights into WMMA B-fragment order:
// per net: [WF0: 2048 x float2][WF1: 2048 x float2][b0: 64][b1: 64]  (8320 floats)
__global__ __launch_bounds__(256)
void born_fold_kernel(const float* __restrict__ eW, const float* __restrict__ eb,
                      const float* __restrict__ eg, const float* __restrict__ ebeta,
                      const float* __restrict__ em, const float* __restrict__ ev,
                      const float* __restrict__ tW, const float* __restrict__ tb,
                      const float* __restrict__ tg, const float* __restrict__ tbeta,
                      const float* __restrict__ tm, const float* __restrict__ tv,
                      float* __restrict__ wf)
{
    const int t = blockIdx.x * blockDim.x + threadIdx.x;
    if (t >= 256) return;
    const int net   = t >> 7;          // 0 = edge, 1 = triplet
    const int layer = (t >> 6) & 1;
    const int col   = t & 63;
    const float* W    = net ? tW    : eW;
    const float* b    = net ? tb    : eb;
    const float* g    = net ? tg    : eg;
    const float* beta = net ? tbeta : ebeta;
    const float* m    = net ? tm    : em;
    const float* v    = net ? tv    : ev;
    const int li = layer * 64 + col;
    const float s  = g[li] * rsqrtf(v[li] + BNEPS);
    float* Wd = wf + (size_t)net * 8320 + (size_t)layer * 4096;  // fragment-packed
    float* bd = wf + (size_t)net * 8320 + 8192 + (size_t)layer * 64;
    bd[col] = (b[li] - m[li]) * s + beta[li];
    for (int p = 0; p < 32; ++p) {
        Wd[(p * 64 + col) * 2 + 0] = W[layer * 4096 + (2 * p + 0) * 64 + col] * s;
        Wd[(p * 64 + col) * 2 + 1] = W[layer * 4096 + (2 * p + 1) * 64 + col] * s;
    }
}

__global__ __launch_bounds__(256)
void born_geom_kernel(const float* __restrict__ pos,
                      const float* __restrict__ shift,
                      const int* __restrict__ edge_index, int E,
                      float* __restrict__ dir, float* __restrict__ len)
{
    const int e = blockIdx.x * 256 + threadIdx.x;
    if (e >= E) return;
    const int j = edge_index[e];       // edge_index[0][e]
    const int i = edge_index[E + e];   // edge_index[1][e]
    const float dx = pos[i * 3 + 0] + shift[e * 3 + 0] - pos[j * 3 + 0];
    const float dy = pos[i * 3 + 1] + shift[e * 3 + 1] - pos[j * 3 + 1];
    const float dz = pos[i * 3 + 2] + shift[e * 3 + 2] - pos[j * 3 + 2];
    const float L = sqrtf(dx * dx + dy * dy + dz * dz);
    const float inv = 1.0f / L;
    dir[e * 3 + 0] = dx * inv;
    dir[e * 3 + 1] = dy * inv;
    dir[e * 3 + 2] = dz * inv;
    len[e] = L;
}

__global__ __launch_bounds__(256)
void born_zero_kernel(float* __restrict__ out, int n)
{
    const int i = blockIdx.x * 256 + threadIdx.x;
    if (i < n) out[i] = 0.0f;
}

__global__ __launch_bounds__(256)
void born_edge_kernel(const float* __restrict__ edge_attr,
                      const v2f* __restrict__ WF0, const float* __restrict__ b0,
                      const v2f* __restrict__ WF1, const float* __restrict__ b1,
                      const float* __restrict__ Wo, const float* __restrict__ bo,
                      const float* __restrict__ dir,
                      const int* __restrict__ edge_index, int E,
                      float* __restrict__ born)
{
    __shared__ float sRow[32];
    const int row0 = blockIdx.x * 32;
    mlp_core(edge_attr, WF0, b0, WF1, b1, Wo, bo, row0, E, sRow);
    for (int idx = threadIdx.x; idx < 288; idx += 256) {
        const int r = idx / 9, c = idx % 9;
        const int e = row0 + r;
        if (e < E) {
            const int i_node = edge_index[E + e];
            const float val = sRow[r] * dir[e * 3 + c / 3] * dir[e * 3 + c % 3];
            atomicAdd(&born[(size_t)i_node * 9 + c], val);
        }
    }
}

__global__ __launch_bounds__(256)
void born_trip_kernel(const float* __restrict__ trip_attr,
                      const v2f* __restrict__ WF0, const float* __restrict__ b0,
                      const v2f* __restrict__ WF1, const float* __restrict__ b1,
                      const float* __restrict__ Wo, const float* __restrict__ bo,
                      const float* __restrict__ dir, const float* __restrict__ len,
                      const int* __restrict__ idx_j,
                      const int* __restrict__ idx_kj,
                      const int* __restrict__ idx_ji, int T,
                      float* __restrict__ born)
{
    __shared__ float sRow[32];
    const int row0 = blockIdx.x * 32;
    mlp_core(trip_attr, WF0, b0, WF1, b1, Wo, bo, row0, T, sRow);
    for (int idx = threadIdx.x; idx < 288; idx += 256) {
        const int r = idx / 9, c = idx % 9;
        const int t = row0 + r;
        if (t < T) {
            const int kj = idx_kj[t];
            const int ji = idx_ji[t];
            const float msk = (len[kj] < TCUT && len[ji] < TCUT) ? 1.0f : 0.0f;
            const float val = sRow[r] * msk * dir[kj * 3 + c / 3] * dir[ji * 3 + c % 3];
            atomicAdd(&born[(size_t)idx_j[t] * 9 + c], val);
        }
    }
}

// ---------------- launcher ----------------

extern "C" void kernel_launch(void* const* d_in, const int* in_sizes, int n_in,
                              void* d_out, int out_size, void* d_ws, size_t ws_size,
                              hipStream_t stream)
{
    const float* pos       = (const float*)d_in[0];
    const float* nbr_shift = (const float*)d_in[1];
    const float* edge_attr = (const float*)d_in[2];
    const float* trip_attr = (const float*)d_in[3];
    const float* eW    = (const float*)d_in[4];
    const float* eb    = (const float*)d_in[5];
    const float* eg    = (const float*)d_in[6];
    const float* ebeta = (const float*)d_in[7];
    const float* em    = (const float*)d_in[8];
    const float* ev    = (const float*)d_in[9];
    const float* eWo   = (const float*)d_in[10];
    const float* ebo   = (const float*)d_in[11];
    const float* tW    = (const float*)d_in[12];
    const float* tb    = (const float*)d_in[13];
    const float* tg    = (const float*)d_in[14];
    const float* tbeta = (const float*)d_in[15];
    const float* tm    = (const float*)d_in[16];
    const float* tv    = (const float*)d_in[17];
    const float* tWo   = (const float*)d_in[18];
    const float* tbo   = (const float*)d_in[19];
    const int* edge_index = (const int*)d_in[20];
    const int* idx_j   = (const int*)d_in[21];
    const int* idx_kj  = (const int*)d_in[22];
    const int* idx_ji  = (const int*)d_in[23];

    const int N = in_sizes[0] / 3;
    const int E = in_sizes[1] / 3;
    const int T = in_sizes[21];

    float* ws  = (float*)d_ws;
    float* dir = ws;                       // 3E floats
    float* len = ws + (size_t)3 * E;       // E floats
    float* wf  = ws + (size_t)4 * E;       // 2 nets * 8320 floats (frag-packed)
    const v2f*   WF0e = (const v2f*)(wf);
    const v2f*   WF1e = (const v2f*)(wf + 4096);
    const float* b0e  = wf + 8192;
    const float* b1e  = wf + 8256;
    const v2f*   WF0t = (const v2f*)(wf + 8320);
    const v2f*   WF1t = (const v2f*)(wf + 8320 + 4096);
    const float* b0t  = wf + 8320 + 8192;
    const float* b1t  = wf + 8320 + 8256;
    float* born = (float*)d_out;

    born_fold_kernel<<<1, 256, 0, stream>>>(eW, eb, eg, ebeta, em, ev,
                                            tW, tb, tg, tbeta, tm, tv, wf);
    born_geom_kernel<<<(E + 255) / 256, 256, 0, stream>>>(pos, nbr_shift,
                                                          edge_index, E, dir, len);
    born_zero_kernel<<<(N * 9 + 255) / 256, 256, 0, stream>>>(born, N * 9);

    born_edge_kernel<<<(E + 31) / 32, 256, 0, stream>>>(
        edge_attr, WF0e, b0e, WF1e, b1e, eWo, ebo, dir, edge_index, E, born);

    born_trip_kernel<<<(T + 31) / 32, 256, 0, stream>>>(
        trip_attr, WF0t, b0t, WF1t, b1t, tWo, tbo, dir, len,
        idx_j, idx_kj, idx_ji, T, born);
}